// VanilaGCN_63256278335719
// MI455X (gfx1250) — compile-verified
//
#include <hip/hip_runtime.h>
#include <stdint.h>

typedef __attribute__((ext_vector_type(2))) float v2f;
typedef __attribute__((ext_vector_type(8))) float v8f;

#define NNODES   50000
#define NEDGES   800000
#define FEAT     96
#define NGRAPHS  512
#define NCLASSES 10
#define TM       64          // rows per GEMM block

static inline unsigned cdiv(unsigned a, unsigned b) { return (a + b - 1) / b; }

// ---------------- elementwise / degree ----------------

__global__ void k_fill1(float* __restrict__ p, int n) {
  int i = blockIdx.x * blockDim.x + threadIdx.x;
  if (i < n) p[i] = 1.0f;
}

__global__ void k_deg_accum(const long long* __restrict__ dst,
                            const float* __restrict__ w,
                            float* __restrict__ deg, int E) {
  int e = blockIdx.x * blockDim.x + threadIdx.x;
  if (e < E) atomicAdd(&deg[(int)dst[e]], w[e]);
}

__global__ void k_rsqrt_inplace(float* __restrict__ p, int n) {
  int i = blockIdx.x * blockDim.x + threadIdx.x;
  if (i < n) {
    float d = p[i];
    p[i] = (d > 0.0f) ? rsqrtf(d) : 0.0f;
  }
}

__global__ void k_edge_norm(const long long* __restrict__ src,
                            const long long* __restrict__ dst,
                            const float* __restrict__ w,
                            const float* __restrict__ dis,
                            float* __restrict__ nrm, int E) {
  int e = blockIdx.x * blockDim.x + threadIdx.x;
  if (e < E) nrm[e] = dis[(int)src[e]] * w[e] * dis[(int)dst[e]];
}

// ---------------- fp32 WMMA GEMM: C[M x 96] = A[M x 96] * W[96 x 96] -----------
// Per block: 64 rows x 96 cols. 4 waves, each owns a 16-row strip and sweeps
// six 16-col tiles; K=96 consumed 4 at a time by V_WMMA_F32_16X16X4_F32.
// Tiles are staged with GLOBAL_LOAD_ASYNC_TO_LDS_B128 (ASYNCcnt-tracked),
// writing straight into LDS with no VGPR round-trip.

__global__ __launch_bounds__(128) void k_gemm_wmma(const float* __restrict__ A,
                                                   const float* __restrict__ W,
                                                   float* __restrict__ C, int M) {
  __shared__ float sA[TM][100];   // padded rows (400B stride) -> bank-spread A reads
  __shared__ float sB[96][96];    // W row-major: contiguous async copy target

  const int tid = threadIdx.x;      // 0..127
  const int m0  = blockIdx.x * TM;

  const unsigned ldsA = (unsigned)(size_t)(&sA[0][0]);  // low 32b of generic = LDS offset
  const unsigned ldsB = (unsigned)(size_t)(&sB[0][0]);

  // --- async stage W: 96*96*4 = 36864 B = 2304 x 16B chunks, fully contiguous ---
  #pragma unroll
  for (int j = 0; j < 18; ++j) {
    unsigned c   = (unsigned)tid + 128u * j;   // < 2304
    unsigned off = c * 16u;
    unsigned dst = ldsB + off;
    asm volatile("global_load_async_to_lds_b128 %0, %1, %2"
                 :: "v"(dst), "v"(off), "s"(W) : "memory");
  }

  // --- stage A tile: 64 rows x 24 x 16B chunks ---
  if (m0 + TM <= M) {
    #pragma unroll
    for (int j = 0; j < 12; ++j) {
      unsigned c   = (unsigned)tid + 128u * j;  // < 1536
      unsigned row = c / 24u, cc = c % 24u;
      unsigned dst = ldsA + row * 400u + cc * 16u;
      unsigned off = (unsigned)(m0) * 384u + row * 384u + cc * 16u;
      asm volatile("global_load_async_to_lds_b128 %0, %1, %2"
                   :: "v"(dst), "v"(off), "s"(A) : "memory");
    }
  } else {
    // tail block: guarded scalar staging with zero fill (never reads past A)
    for (int i = tid; i < TM * 96; i += 128) {
      int r = i / 96, c = i % 96;
      int gr = m0 + r;
      sA[r][c] = (gr < M) ? A[(long long)gr * FEAT + c] : 0.0f;
    }
  }

  asm volatile("s_wait_asynccnt 0" ::: "memory");  // each wave drains its own ASYNCcnt
  __syncthreads();                                  // then cross-wave barrier

  const int wave = tid >> 5;          // 0..3 -> 16-row strip
  const int lane = tid & 31;
  const int lm   = lane & 15;         // M index (A) / N index (B,C)
  const int kh   = (lane >> 4) << 1;  // lanes 16..31 hold K+2,K+3
  const int rowBase = wave * 16;

  for (int nt = 0; nt < 6; ++nt) {
    const int n0 = nt * 16;
    v8f acc = {};
    #pragma unroll
    for (int k = 0; k < 96; k += 4) {
      const int k0 = k + kh;
      v2f a, b;
      a.x = sA[rowBase + lm][k0];
      a.y = sA[rowBase + lm][k0 + 1];
      b.x = sB[k0][n0 + lm];
      b.y = sB[k0 + 1][n0 + lm];
      acc = __builtin_amdgcn_wmma_f32_16x16x4_f32(
          /*neg_a=*/false, a, /*neg_b=*/false, b,
          /*c_mod=*/(short)0, acc, /*reuse_a=*/false, /*reuse_b=*/false);
    }
    // C/D layout: lanes 0-15 -> N=lm, M=v ; lanes 16-31 -> N=lm, M=v+8
    const int cn   = n0 + lm;
    const int mAdd = (lane >> 4) * 8;
    #pragma unroll
    for (int v = 0; v < 8; ++v) {
      int gm = m0 + rowBase + mAdd + v;
      if (gm < M) C[(long long)gm * FEAT + cn] = acc[v];
    }
  }
}

// ---------------- GCN aggregation ----------------

// out[i][f] = dis[i]^2 * xw[i][f] + bias[f]   (self-loop term, norm = dis^2)
__global__ void k_node_init(const float* __restrict__ xw,
                            const float* __restrict__ dis,
                            const float* __restrict__ bias,
                            float* __restrict__ out, int n) {
  int i = blockIdx.x * blockDim.x + threadIdx.x;
  if (i < n * FEAT) {
    int node = i / FEAT, f = i % FEAT;
    float d = dis[node];
    out[i] = d * d * xw[i] + bias[f];
  }
}

// wave per edge: out[dst] += norm * xw[src]  (96 floats = 3 x 32 lanes)
__global__ __launch_bounds__(256) void k_scatter(const long long* __restrict__ src,
                                                 const long long* __restrict__ dst,
                                                 const float* __restrict__ nrm,
                                                 const float* __restrict__ xw,
                                                 float* __restrict__ out, int E) {
  int wavesPerBlock = blockDim.x >> 5;
  int e = blockIdx.x * wavesPerBlock + (threadIdx.x >> 5);
  if (e >= E) return;
  int lane = threadIdx.x & 31;
  long long s = src[e], d = dst[e];
  float nv = nrm[e];
  const float* __restrict__ row = xw + s * FEAT;
  float* __restrict__ orow = out + d * FEAT;
  #pragma unroll
  for (int j = 0; j < 3; ++j) {
    int f = lane + 32 * j;
    atomicAdd(&orow[f], nv * row[f]);
  }
}

__global__ void k_relu(float* __restrict__ p, int n) {
  int i = blockIdx.x * blockDim.x + threadIdx.x;
  if (i < n) p[i] = fmaxf(p[i], 0.0f);
}

// ---------------- pooling + head ----------------

__global__ void k_pool_sum(const float* __restrict__ h,
                           const long long* __restrict__ batch,
                           float* __restrict__ sums, int n) {
  int i = blockIdx.x * blockDim.x + threadIdx.x;
  if (i < n * FEAT) {
    int node = i / FEAT, f = i % FEAT;
    atomicAdd(&sums[(int)batch[node] * FEAT + f], h[i]);
  }
}

__global__ void k_pool_cnt(const long long* __restrict__ batch,
                           float* __restrict__ cnt, int n) {
  int i = blockIdx.x * blockDim.x + threadIdx.x;
  if (i < n) atomicAdd(&cnt[(int)batch[i]], 1.0f);
}

__global__ void k_head(const float* __restrict__ sums,
                       const float* __restrict__ cnt,
                       const float* __restrict__ Wl,
                       const float* __restrict__ bl,
                       float* __restrict__ out) {
  int i = blockIdx.x * blockDim.x + threadIdx.x;
  if (i < NGRAPHS * NCLASSES) {
    int g = i / NCLASSES, c = i % NCLASSES;
    float inv = 1.0f / fmaxf(cnt[g], 1.0f);
    float acc = 0.0f;
    #pragma unroll
    for (int f = 0; f < FEAT; ++f)
      acc = fmaf(sums[g * FEAT + f] * inv, Wl[f * NCLASSES + c], acc);
    out[i] = acc + bl[c];
  }
}

// ---------------- launcher ----------------

extern "C" void kernel_launch(void* const* d_in, const int* in_sizes, int n_in,
                              void* d_out, int out_size, void* d_ws, size_t ws_size,
                              hipStream_t stream) {
  const float*     x    = (const float*)d_in[0];
  const float*     ew   = (const float*)d_in[1];   // [E,1]
  const float*     W1   = (const float*)d_in[2];
  const float*     b1   = (const float*)d_in[3];
  const float*     W2   = (const float*)d_in[4];
  const float*     b2   = (const float*)d_in[5];
  const float*     W3   = (const float*)d_in[6];
  const float*     b3   = (const float*)d_in[7];
  const float*     Wl   = (const float*)d_in[8];
  const float*     bl   = (const float*)d_in[9];
  const long long* ei   = (const long long*)d_in[10];  // [2,E] int64
  const long long* bat  = (const long long*)d_in[11];  // [N] int64
  float* out = (float*)d_out;

  const long long* src = ei;
  const long long* dst = ei + NEDGES;

  float* ws   = (float*)d_ws;
  float* dis  = ws;                                   // NNODES (deg, then deg^-1/2)
  float* nrm  = dis + NNODES;                         // NEDGES
  float* xw   = nrm + NEDGES;                         // NNODES*FEAT
  float* hA   = xw + (size_t)NNODES * FEAT;           // NNODES*FEAT
  float* hB   = hA + (size_t)NNODES * FEAT;           // NNODES*FEAT
  float* sums = hB + (size_t)NNODES * FEAT;           // NGRAPHS*FEAT
  float* cnt  = sums + (size_t)NGRAPHS * FEAT;        // NGRAPHS

  const int BS = 256;
  const unsigned gN   = cdiv(NNODES, BS);
  const unsigned gE   = cdiv(NEDGES, BS);
  const unsigned gNF  = cdiv(NNODES * FEAT, BS);
  const unsigned gGem = cdiv(NNODES, TM);
  const unsigned gSct = cdiv(NEDGES, BS / 32);        // 8 edges per 256-thread block

  // --- gcn_norm (shared across all 3 layers) ---
  k_fill1<<<gN, BS, 0, stream>>>(dis, NNODES);                       // self-loop weight 1
  k_deg_accum<<<gE, BS, 0, stream>>>(dst, ew, dis, NEDGES);
  k_rsqrt_inplace<<<gN, BS, 0, stream>>>(dis, NNODES);
  k_edge_norm<<<gE, BS, 0, stream>>>(src, dst, ew, dis, nrm, NEDGES);

  // --- layer 1 ---
  k_gemm_wmma<<<gGem, 128, 0, stream>>>(x, W1, xw, NNODES);
  k_node_init<<<gNF, BS, 0, stream>>>(xw, dis, b1, hA, NNODES);
  k_scatter<<<gSct, BS, 0, stream>>>(src, dst, nrm, xw, hA, NEDGES);
  k_relu<<<gNF, BS, 0, stream>>>(hA, NNODES * FEAT);

  // --- layer 2 ---
  k_gemm_wmma<<<gGem, 128, 0, stream>>>(hA, W2, xw, NNODES);
  k_node_init<<<gNF, BS, 0, stream>>>(xw, dis, b2, hB, NNODES);
  k_scatter<<<gSct, BS, 0, stream>>>(src, dst, nrm, xw, hB, NEDGES);
  k_relu<<<gNF, BS, 0, stream>>>(hB, NNODES * FEAT);

  // --- layer 3 (no relu) ---
  k_gemm_wmma<<<gGem, 128, 0, stream>>>(hB, W3, xw, NNODES);
  k_node_init<<<gNF, BS, 0, stream>>>(xw, dis, b3, hA, NNODES);
  k_scatter<<<gSct, BS, 0, stream>>>(src, dst, nrm, xw, hA, NEDGES);

  // --- global mean pool + linear head ---
  hipMemsetAsync(sums, 0, (size_t)(NGRAPHS * FEAT + NGRAPHS) * sizeof(float), stream);
  k_pool_sum<<<gNF, BS, 0, stream>>>(hA, bat, sums, NNODES);
  k_pool_cnt<<<gN, BS, 0, stream>>>(bat, cnt, NNODES);
  k_head<<<cdiv(NGRAPHS * NCLASSES, BS), BS, 0, stream>>>(sums, cnt, Wl, bl, out);
}